// Cross_Attention_6485400617372
// MI455X (gfx1250) — compile-verified
//
#include <hip/hip_runtime.h>

#define D     128
#define NSEQ  2048
#define BATCH 8

typedef __attribute__((ext_vector_type(2))) float v2f;
typedef __attribute__((ext_vector_type(4))) float v4f;
typedef __attribute__((ext_vector_type(8))) float v8f;
typedef int v4i_gcc __attribute__((vector_size(16)));   // matches builtin proto

// ---------------------------------------------------------------------------
// Stage a contiguous 16x128 fp32 tile (2048 floats) into LDS.
// Preferred path: CDNA5 async global->LDS b128 (ASYNCcnt) + s_wait_asynccnt.
// Fallback: plain vector copy through VGPRs.
// ---------------------------------------------------------------------------
__device__ __forceinline__ void stage_tile_2048(float* __restrict__ dstLds,
                                                const float* __restrict__ srcGlobal,
                                                int tid) {
#if __has_builtin(__builtin_amdgcn_global_load_async_to_lds_b128)
  v4i_gcc* gp = (v4i_gcc*)(srcGlobal + tid * 4);   // global source (16B/lane)
  v4i_gcc* lp = (v4i_gcc*)(dstLds + tid * 4);      // LDS destination
  // 256 lanes x 16B = 4 KB per op; INST_OFFSET applies to both LDS and memory
  __builtin_amdgcn_global_load_async_to_lds_b128(gp, lp, 0, 0);
  __builtin_amdgcn_global_load_async_to_lds_b128(gp, lp, 4096, 0);
#if __has_builtin(__builtin_amdgcn_s_wait_asynccnt)
  __builtin_amdgcn_s_wait_asynccnt(0);
#else
  asm volatile("s_wait_asynccnt 0" ::: "memory");
#endif
#else
  ((v4f*)dstLds)[tid]       = ((const v4f*)srcGlobal)[tid];
  ((v4f*)dstLds)[tid + 256] = ((const v4f*)srcGlobal)[tid + 256];
#endif
}

// ---------------------------------------------------------------------------
// Kernel 1: linear projection  out[r][d] = sum_k x[r][k]*W[d][k] + bias[d]
// grid.x = R/16 row tiles; block = 256 (8 waves); wave w -> cols w*16..
// ---------------------------------------------------------------------------
__global__ __launch_bounds__(256)
void proj_kernel(const float* __restrict__ x, const float* __restrict__ W,
                 const float* __restrict__ bias, float* __restrict__ out) {
  __shared__ __align__(16) float xs[16 * D];
  const int tid  = threadIdx.x;
  const int wave = tid >> 5;
  const int lane = tid & 31;
  const int l = lane & 15;
  const int h = lane >> 4;

  stage_tile_2048(xs, x + (long)blockIdx.x * 16 * D, tid);
  __syncthreads();

  const int ct = wave * 16;
  v8f acc = {};
#pragma unroll
  for (int kc = 0; kc < D / 4; ++kc) {
    const int k0 = kc * 4 + h * 2;
    v2f a  = *(const v2f*)(xs + l * D + k0);           // A: M=l, K=k0(+2h)+v
    v2f bf = *(const v2f*)(W + (ct + l) * D + k0);     // B: N=l, K row = k0+p
    acc = __builtin_amdgcn_wmma_f32_16x16x4_f32(false, a, false, bf,
                                                (short)0, acc, false, false);
  }
  const float bb = bias[ct + l];
  const long orow = (long)blockIdx.x * 16;
#pragma unroll
  for (int p = 0; p < 8; ++p) {
    out[(orow + h * 8 + p) * D + ct + l] = acc[p] + bb;   // C/D: M=8h+p, N=l
  }
}

// ---------------------------------------------------------------------------
// Kernel 2 (fused): logits + exact 1.5-entmax, logits never touch HBM.
// One block = 16 query rows. Phase 1: 16x2048 scaled logits into 128KB LDS
// (8 waves x 16 col-tiles x 32 fp32 WMMA). Phase 2: per-row bisection for tau
// (16 threads/row, shfl_xor reductions), single coalesced output write.
// Scale 0.5/sqrt(D) folds the entmax z/2; after rowmax shift tau* in [-1,0].
// ---------------------------------------------------------------------------
__global__ __launch_bounds__(256)
void attn_entmax_kernel(const float* __restrict__ Q, const float* __restrict__ K,
                        float* __restrict__ out) {
  extern __shared__ __align__(16) float lds[];
  float* S  = lds;               // 16 * NSEQ logits panel
  float* qs = lds + 16 * NSEQ;   // 16 * D staged Q tile

  const int tid = threadIdx.x, wave = tid >> 5, lane = tid & 31;
  const int l = lane & 15, h = lane >> 4;
  const int b  = blockIdx.x / (NSEQ / 16);
  const int rt = (blockIdx.x % (NSEQ / 16)) * 16;

  const float* Qb = Q + (long)b * NSEQ * D;
  const float* Kb = K + (long)b * NSEQ * D;

  stage_tile_2048(qs, Qb + (long)rt * D, tid);
  __syncthreads();

  const float scale = 0.044194173824159223f;   // 0.5/sqrt(128)

  // ---- Phase 1: logits panel ----
  for (int j = 0; j < NSEQ / 128; ++j) {       // 16 column groups of 128
    const int ct = j * 128 + wave * 16;
    const float* krow = Kb + (long)(ct + l) * D;
    __builtin_prefetch(krow, 0, 1);            // global_prefetch_b8
    v8f acc = {};
#pragma unroll
    for (int kc = 0; kc < D / 4; ++kc) {
      const int k0 = kc * 4 + h * 2;
      v2f a  = *(const v2f*)(qs + l * D + k0);   // ds_load_b64
      v2f bf = *(const v2f*)(krow + k0);         // global_load_b64
      acc = __builtin_amdgcn_wmma_f32_16x16x4_f32(false, a, false, bf,
                                                  (short)0, acc, false, false);
    }
#pragma unroll
    for (int p = 0; p < 8; ++p) {
      S[(h * 8 + p) * NSEQ + ct + l] = acc[p] * scale;   // ds_store_b32
    }
  }
  __syncthreads();

  // ---- Phase 2: entmax per row (16 threads per row) ----
  const int row = tid >> 4;        // 0..15
  const int g   = tid & 15;        // lane within row group
  float* Srow = S + row * NSEQ;

  float m = -3.0e38f;
  for (int i = g; i < NSEQ; i += 16) m = fmaxf(m, Srow[i]);
#pragma unroll
  for (int off = 8; off; off >>= 1) m = fmaxf(m, __shfl_xor(m, off, 16));

  for (int i = g; i < NSEQ; i += 16) Srow[i] -= m;   // shift: max(z)=0

  float lo = -1.0f, hi = 0.0f;
  for (int it = 0; it < 26; ++it) {
    const float tau = 0.5f * (lo + hi);
    float part = 0.0f;
    for (int i = g; i < NSEQ; i += 16) {
      float t = fmaxf(Srow[i] - tau, 0.0f);
      part += t * t;
    }
#pragma unroll
    for (int off = 8; off; off >>= 1) part += __shfl_xor(part, off, 16);
    if (part > 1.0f) lo = tau; else hi = tau;        // f decreasing in tau
  }
  const float tau = 0.5f * (lo + hi);

  float* orow = out + ((long)b * NSEQ + rt + row) * NSEQ;
  for (int i = g; i < NSEQ; i += 16) {
    float t = fmaxf(Srow[i] - tau, 0.0f);
    orow[i] = t * t;
  }
}

// ---------------------------------------------------------------------------
extern "C" void kernel_launch(void* const* d_in, const int* in_sizes, int n_in,
                              void* d_out, int out_size, void* d_ws, size_t ws_size,
                              hipStream_t stream) {
  const float* x_c = (const float*)d_in[0];
  const float* x_n = (const float*)d_in[1];
  const float* Wq  = (const float*)d_in[2];
  const float* bq  = (const float*)d_in[3];
  const float* Wk  = (const float*)d_in[4];
  const float* bk  = (const float*)d_in[5];
  float* out = (float*)d_out;

  float* Qw = (float*)d_ws;                       //  8 MB
  float* Kw = Qw + (size_t)BATCH * NSEQ * D;      // +8 MB

  const int rowTiles = (BATCH * NSEQ) / 16;       // 1024
  proj_kernel<<<rowTiles, 256, 0, stream>>>(x_c, Wq, bq, Qw);
  proj_kernel<<<rowTiles, 256, 0, stream>>>(x_n, Wk, bk, Kw);

  // 16*NSEQ logits + 16*D Q tile, fp32
  const size_t ldsBytes = (size_t)(16 * NSEQ + 16 * D) * sizeof(float);  // 136 KB
  attn_entmax_kernel<<<BATCH * (NSEQ / 16), 256, ldsBytes, stream>>>(Qw, Kw, out);
}